// Model_60060822667341
// MI455X (gfx1250) — compile-verified
//
#include <hip/hip_runtime.h>
#include <cmath>

typedef float v2f __attribute__((ext_vector_type(2)));
typedef float v8f __attribute__((ext_vector_type(8)));

#define HL 1023      // history length
#define HT 128       // time slots
#define HLAMBDA 10.0f

#define LOG2E 1.4426950408889634f
#define LN2   0.6931471805599453f

// One workgroup, 1024 threads (32 waves).
// Phase 1: Hillis-Steele scan of affine maps -> E_i in LDS (O(L)).
// Phase 2: waves 0..7 compute C2_t = sum_i log2(mu_t + alpha_t*E_i) via
//          V_WMMA_F32_16X16X4_F32 with B = ones (row-sum in matrix pipe).
//          (final *ln2 applied once per row; raw v_log_f32, no libm fixups)
// Phase 3: threads 0..127 finish the closed-form terms.
__global__ __launch_bounds__(1024) void hawkes_ll_kernel(
    const float* __restrict__ history,     // (HL)
    const float* __restrict__ next_time,   // (HT)
    const float* __restrict__ mu,          // (HT)
    const float* __restrict__ alpha,       // (HT)
    const float* __restrict__ omega_p,     // (1)
    float* __restrict__ out)               // (HT)
{
    __shared__ float sA[1024];
    __shared__ float sB[1024];
    __shared__ float sE[1024];
    __shared__ float sC[HT];

    const int tid = threadIdx.x;
    const float omega = omega_p[0];
    const float wl2 = omega * LOG2E;       // exp(-w*d) == exp2(-wl2*d)

    // ---- Phase 1: scan  E_i = a_i*(E_{i-1}+1),  a_i = exp(-w*(h_i-h_{i-1}))
    // Represent prefix as affine map E_i = A*0 + B; compose (A,B) pairs.
    {
        float a = 0.0f, b = 0.0f;
        if (tid > 0 && tid < HL) {
            float ai = __builtin_amdgcn_exp2f(-wl2 * (history[tid] - history[tid - 1]));
            a = ai; b = ai;              // f_i(x) = a_i*x + a_i   (E_0 = 0)
        }
        sA[tid] = a; sB[tid] = b;
        __syncthreads();
        for (int d = 1; d < 1024; d <<= 1) {
            float ai = sA[tid], bi = sB[tid];
            float na = ai, nb = bi;
            if (tid >= d) {
                float ap = sA[tid - d], bp = sB[tid - d];
                na = ai * ap;            // (f_i o f_prev): A = A_i*A_p
                nb = ai * bp + bi;       //                 B = A_i*B_p + B_i
            }
            __syncthreads();
            sA[tid] = na; sB[tid] = nb;
            __syncthreads();
        }
        sE[tid] = (tid < HL) ? sB[tid] : 0.0f;   // pad E[1023] = 0
        __syncthreads();
    }

    // ---- Phase 2: C2_t = sum_{i<1024} log2(mu_t + alpha_t * E_i) via WMMA
    const int wave = tid >> 5;
    const int lane = tid & 31;
    if (wave < 8) {
        const int row = (wave << 4) + (lane & 15);   // t handled by this lane
        const float mt = mu[row];
        const float at = alpha[row];
        const int kh = (lane >> 4) << 1;             // lanes 0-15 -> K0,K1; 16-31 -> K2,K3
        v2f ones; ones.x = 1.0f; ones.y = 1.0f;      // B = 4x16 all-ones
        v8f acc = {};                                // 16x16 f32 accumulator
        for (int i0 = 0; i0 < 1024; i0 += 4) {
            v2f av;
            av.x = __builtin_amdgcn_logf(fmaf(at, sE[i0 + kh],     mt));
            av.y = __builtin_amdgcn_logf(fmaf(at, sE[i0 + kh + 1], mt));
            acc = __builtin_amdgcn_wmma_f32_16x16x4_f32(
                false, av, false, ones, (short)0, acc, false, false);
        }
        // D layout: VGPR r, lane 0 -> (M=r, N=0); lane 16 -> (M=8+r, N=0).
        if ((lane & 15) == 0) {
            const int tbase = (wave << 4) + ((lane >> 4) << 3);
            #pragma unroll
            for (int r = 0; r < 8; ++r) sC[tbase + r] = acc[r];
        }
    }
    __syncthreads();

    // ---- Phase 3: closed-form remainder per time slot
    if (tid < HT) {
        const float nt = next_time[tid];
        const float mt = mu[tid];
        const float at = alpha[tid];
        const float Elast = sE[HL - 1];
        const float hlast = history[HL - 1];
        // S_t = sum_j exp(-w*(nt - h_j)) = exp(-w*(nt - h_last)) * (E_last + 1)
        const float S = __builtin_amdgcn_exp2f(-wl2 * (nt - hlast)) * (Elast + 1.0f);
        // remove the pad term log2(mu + alpha*0), add the next-event row; *ln2 once
        const float summed = LN2 * (sC[tid] - __builtin_amdgcn_logf(mt)
                                   + __builtin_amdgcn_logf(fmaf(at, S, mt)));
        const float alpha_term = (at / omega) * ((float)HL - S);
        const float value = summed - alpha_term - mt * nt;
        out[tid] = -value + (at * at + mt * mt) * HLAMBDA;
    }
}

extern "C" void kernel_launch(void* const* d_in, const int* in_sizes, int n_in,
                              void* d_out, int out_size, void* d_ws, size_t ws_size,
                              hipStream_t stream) {
    (void)in_sizes; (void)n_in; (void)d_ws; (void)ws_size; (void)out_size;
    const float* history = (const float*)d_in[0];
    const float* next_t  = (const float*)d_in[1];
    const float* mu      = (const float*)d_in[2];
    const float* alpha   = (const float*)d_in[3];
    const float* omega   = (const float*)d_in[4];
    float* out = (float*)d_out;
    hawkes_ll_kernel<<<1, 1024, 0, stream>>>(history, next_t, mu, alpha, omega, out);
}